// SPDTangentSpace_13546326851715
// MI455X (gfx1250) — compile-verified
//
#include <hip/hip_runtime.h>
#include <math.h>

typedef float v2f __attribute__((ext_vector_type(2)));
typedef float v8f __attribute__((ext_vector_type(8)));

#define NDIM 64
#define TPB  128
// XOR-swizzled LDS address: row-major 64-wide, column xor'd with row.
// Bijective per row; diagonal (r,r) maps to column 0 -> cheap +/- I updates,
// and A-fragment column loads (16 lanes, same column, different rows) hit
// 16 distinct banks instead of one.
#define SWZ(r, c) (((r) << 6) | ((c) ^ (r)))

// GEMM epilogues (fused affine transforms of the accumulator):
//   EPI_NONE: d = acc
//   EPI_NS  : d = 1.5*I - 0.5*acc      (Newton-Schulz T = 1.5I - 0.5*Z*Y)
//   EPI_DIAG: d = acc + cj*I           (Horner T = cj*I + E*T)
#define EPI_NONE 0
#define EPI_NS   1
#define EPI_DIAG 2

// D = epilogue(A * B): 64x64x64 fp32 GEMM on one workgroup (4 waves, wave32).
// 4x4 grid of 16x16 C-tiles; wave w owns the 2x2 tile block at
// (bi0 = (w>>1)*2, bj0 = (w&1)*2). Each A/B fragment feeds 2 WMMAs ->
// 8 ds_load_b32 per 4 WMMAs (2 loads/WMMA).
// Caller guarantees D does not alias A or B, and that A/B writers have
// barriered before the call; mm64 ends with a single __syncthreads().
template <int EPI>
__device__ __forceinline__ void mm64(const float* A, const float* B, float* D,
                                     int wave, int lane, float cj = 0.0f) {
  const int half = lane >> 4;            // 0: lanes 0-15, 1: lanes 16-31
  const int l    = lane & 15;
  const int bi0  = (wave >> 1) * 2;      // C-tile row blocks bi0, bi0+1
  const int bj0  = (wave & 1) * 2;       // C-tile col blocks bj0, bj0+1
  const int ar0  = bi0 * 16 + l;         // A fragment row, tile-row 0
  const int ar1  = ar0 + 16;             // A fragment row, tile-row 1
  const int c0   = bj0 * 16 + l;         // B/C column, tile-col 0
  const int c1   = c0 + 16;              // B/C column, tile-col 1

  v8f acc00 = {};
  v8f acc01 = {};
  v8f acc10 = {};
  v8f acc11 = {};
  for (int k = 0; k < NDIM; k += 4) {
    const int ka = k + 2 * half;         // K pair handled by this half-wave
    v2f a0, a1, b0, b1;
    a0.x = A[SWZ(ar0, ka)];
    a0.y = A[SWZ(ar0, ka + 1)];
    a1.x = A[SWZ(ar1, ka)];
    a1.y = A[SWZ(ar1, ka + 1)];
    b0.x = B[SWZ(ka,     c0)];
    b0.y = B[SWZ(ka + 1, c0)];
    b1.x = B[SWZ(ka,     c1)];
    b1.y = B[SWZ(ka + 1, c1)];
    acc00 = __builtin_amdgcn_wmma_f32_16x16x4_f32(false, a0, false, b0,
                                                  (short)0, acc00, false, false);
    acc01 = __builtin_amdgcn_wmma_f32_16x16x4_f32(false, a0, false, b1,
                                                  (short)0, acc01, false, false);
    acc10 = __builtin_amdgcn_wmma_f32_16x16x4_f32(false, a1, false, b0,
                                                  (short)0, acc10, false, false);
    acc11 = __builtin_amdgcn_wmma_f32_16x16x4_f32(false, a1, false, b1,
                                                  (short)0, acc11, false, false);
  }
  // C/D layout: VGPR r -> row base + r (lanes 0-15) / base + 8 + r (lanes 16-31)
  const int rb0 = bi0 * 16 + half * 8;
  const int rb1 = rb0 + 16;
#pragma unroll
  for (int r = 0; r < 8; ++r) {
    const int r0 = rb0 + r, r1 = rb1 + r;
    float v00 = acc00[r], v01 = acc01[r], v10 = acc10[r], v11 = acc11[r];
    if (EPI == EPI_NS) {
      v00 = ((r0 == c0) ? 1.5f : 0.0f) - 0.5f * v00;
      v01 = ((r0 == c1) ? 1.5f : 0.0f) - 0.5f * v01;
      v10 = ((r1 == c0) ? 1.5f : 0.0f) - 0.5f * v10;
      v11 = ((r1 == c1) ? 1.5f : 0.0f) - 0.5f * v11;
    } else if (EPI == EPI_DIAG) {
      if (r0 == c0) v00 += cj;
      if (r0 == c1) v01 += cj;
      if (r1 == c0) v10 += cj;
      if (r1 == c1) v11 += cj;
    }
    D[SWZ(r0, c0)] = v00;
    D[SWZ(r0, c1)] = v01;
    D[SWZ(r1, c0)] = v10;
    D[SWZ(r1, c1)] = v11;
  }
  __syncthreads();  // D visible; doubles as entry barrier of the next phase
}

__global__ __launch_bounds__(TPB)
void spd_logmap_kernel(const float* __restrict__ X, float* __restrict__ out) {
  // 4 matrix buffers * 64*64 floats = 64 KB static LDS exactly.
  __shared__ float bufY[NDIM * NDIM];
  __shared__ float bufZ[NDIM * NDIM];
  __shared__ float bufP[NDIM * NDIM];
  __shared__ float bufQ[NDIM * NDIM];
  float* y = bufY;
  float* z = bufZ;
  float* p = bufP;
  float* q = bufQ;

  const int tid  = threadIdx.x;
  const int wave = tid >> 5;
  const int lane = tid & 31;
  const float* xin = X + (size_t)blockIdx.x * (NDIM * NDIM);

  // ---- Load X into LDS (swizzled) + Frobenius-norm reduction (into bufP[0]) ----
  if (tid == 0) p[0] = 0.f;
  __syncthreads();
  float ss = 0.f;
  for (int idx = tid; idx < NDIM * NDIM; idx += TPB) {
    const float v = xin[idx];
    const int r = idx >> 6, c = idx & 63;
    y[SWZ(r, c)] = v;
    ss += v * v;
  }
  for (int off = 16; off > 0; off >>= 1) ss += __shfl_down(ss, off, 32);
  if (lane == 0) atomicAdd(&p[0], ss);
  __syncthreads();
  const float fro   = sqrtf(p[0]);
  const float inv_c = 1.0f / fro;     // c = ||X||_F >= lambda_max
  const float logc  = logf(fro);
  __syncthreads();

  // Y = X / c  (uniform scale: raw-order scan is fine under the swizzle)
  for (int idx = tid; idx < NDIM * NDIM; idx += TPB) y[idx] *= inv_c;
  __syncthreads();

  // ---- 5 square-root stages: Y <- Y^(1/2) via coupled Newton-Schulz ----
  // ||I - X/c||: 0.95, 0.77, 0.52, 0.31, 0.17 per stage; iteration counts
  // sized from quadratic convergence r^(2^n) < 1e-7.
  for (int s = 0; s < 5; ++s) {
    const int niter = (s == 0) ? 9 : (s == 1) ? 7 : (s == 2) ? 6 : (s == 3) ? 5 : 4;
    // Z = I   (diagonal lives at (idx & 63) == 0 under the swizzle)
    for (int idx = tid; idx < NDIM * NDIM; idx += TPB)
      z[idx] = ((idx & 63) == 0) ? 1.0f : 0.0f;
    __syncthreads();
    for (int it = 0; it < niter; ++it) {
      mm64<EPI_NS>(z, y, p, wave, lane);            // P = 1.5I - 0.5*(Z*Y)  (fused)
      mm64<EPI_NONE>(y, p, q, wave, lane);          // Q = Y*P      (new Y)
      mm64<EPI_NONE>(p, z, y, wave, lane);          // Ybuf = P*Z   (new Z; old Y dead)
      float* t = z; z = y; y = q; q = t;            // rotate buffers
    }
  }

  // ---- log(I+E) Taylor/Horner, E = Y - I, 8 terms ----
  for (int idx = tid; idx < NDIM * NDIM; idx += TPB) {
    if ((idx & 63) == 0) y[idx] -= 1.0f;            // E = Y - I
    p[idx] = ((idx & 63) == 0) ? (-1.0f / 8.0f) : 0.0f;  // T = c8 * I
  }
  __syncthreads();
  for (int j = 7; j >= 1; --j) {
    const float cj = ((j & 1) ? 1.0f : -1.0f) / (float)j;
    mm64<EPI_DIAG>(y, p, q, wave, lane, cj);        // T = cj*I + E*T  (fused)
    float* t = p; p = q; q = t;
  }
  mm64<EPI_NONE>(y, p, q, wave, lane);              // Q = E*T = log(I+E)

  // ---- log X = 32*Q + log(c)*I ; emit upper triangle (i<=j) ----
  float* o = out + (size_t)blockIdx.x * 2080;
  for (int t = tid; t < 2080; t += TPB) {
    // invert t -> (i, j): rowstart(i) = i*64 - i*(i-1)/2
    int i = (int)floorf((129.0f - sqrtf(16641.0f - 8.0f * (float)t)) * 0.5f);
    if (i < 0) i = 0;
    if (i > 63) i = 63;
    while (i > 0 && (i * 64 - (i * (i - 1)) / 2) > t) --i;
    while (((i + 1) * 64 - ((i + 1) * i) / 2) <= t) ++i;
    const int j = i + (t - (i * 64 - (i * (i - 1)) / 2));
    float v = 32.0f * q[SWZ(i, j)];
    if (i == j) v += logc;
    o[t] = v;
  }
}

extern "C" void kernel_launch(void* const* d_in, const int* in_sizes, int n_in,
                              void* d_out, int out_size, void* d_ws, size_t ws_size,
                              hipStream_t stream) {
  (void)n_in; (void)out_size; (void)d_ws; (void)ws_size;
  const float* x = (const float*)d_in[0];
  float* out = (float*)d_out;
  const int nmat = in_sizes[0] / (NDIM * NDIM);   // 8192
  spd_logmap_kernel<<<nmat, TPB, 0, stream>>>(x, out);
}